// CRF_34505767256956
// MI455X (gfx1250) — compile-verified
//
#include <hip/hip_runtime.h>
#include <hip/hip_bf16.h>
#include <math.h>

typedef __attribute__((ext_vector_type(2))) float v2f;
typedef __attribute__((ext_vector_type(8))) float v8f;

#define BATCH 64
#define SEQ   512
#define EMB   300
#define TAGS  16
#define ROWS  (BATCH * SEQ)   // 32768 flattened tokens

// -------------------------------------------------------------------------
// Kernel 1: probs = embed_table[text] @ W + b   via V_WMMA_F32_16X16X4_F32
// One wave computes one 16-row x 16-col output tile; K=300 in chunks of 4.
// 8 waves / block -> 128 rows per block -> 256 blocks.
// -------------------------------------------------------------------------
__global__ __launch_bounds__(256) void crf_gemm_kernel(
    const int*   __restrict__ text,
    const float* __restrict__ table,
    const float* __restrict__ W,
    const float* __restrict__ bias,
    float*       __restrict__ probs)
{
    const int lane = threadIdx.x & 31;
    const int wave = threadIdx.x >> 5;
    const int tile = blockIdx.x * 8 + wave;   // 16-row tile index
    const int row0 = tile * 16;

    const int rA   = lane & 15;   // A row (lanes<16) / output column (all lanes)
    const int half = lane >> 4;   // K-half selector for A/B fragments

    // Gather: this lane's A rows come from embedding row of token text[row0+rA]
    const int tok = text[row0 + rA];
    const float* arow = table + (long)tok * EMB;

    v8f c = {};
    for (int k0 = 0; k0 < EMB; k0 += 4) {
        const int ka = k0 + 2 * half;
        // A fragment: lane<16 -> {K0,K1}; lane>=16 -> {K2,K3}; contiguous 8B load
        v2f a = *reinterpret_cast<const v2f*>(arow + ka);
        // B fragment: same K split, column = rA; W is row-major [300][16]
        v2f bm;
        bm.x = W[ka * TAGS + rA];
        bm.y = W[(ka + 1) * TAGS + rA];
        // 8 args: (neg_a, A, neg_b, B, c_mod, C, reuse_a, reuse_b)
        c = __builtin_amdgcn_wmma_f32_16x16x4_f32(
                false, a, false, bm, (short)0, c, false, false);
    }

    // Every element this lane holds is output column rA -> one bias scalar
    const float bv = bias[rA];
    float* outp = probs + (long)(row0 + 8 * half) * TAGS + rA;
#pragma unroll
    for (int r = 0; r < 8; ++r)
        outp[r * TAGS] = c[r] + bv;   // row = row0 + r + 8*half, col = rA
}

// -------------------------------------------------------------------------
// Kernel 2: text_lens[b] = sum(text[b, :] != 0). One block per batch row.
// Writes float copy into d_out and int copy into workspace for kernel 3.
// -------------------------------------------------------------------------
__global__ __launch_bounds__(256) void crf_lens_kernel(
    const int* __restrict__ text,
    float*     __restrict__ lens_f,
    int*       __restrict__ lens_i)
{
    __shared__ int cnt;
    const int b = blockIdx.x;
    if (threadIdx.x == 0) cnt = 0;
    __syncthreads();
    int local = 0;
    for (int t = threadIdx.x; t < SEQ; t += 256)
        local += (text[b * SEQ + t] != 0) ? 1 : 0;
    atomicAdd(&cnt, local);
    __syncthreads();
    if (threadIdx.x == 0) {
        lens_f[b] = (float)cnt;
        lens_i[b] = cnt;
    }
}

// -------------------------------------------------------------------------
// Kernel 3: CRF log-likelihood. One wave per batch row.
// Phase 1: unary + binary scores (all 32 lanes strided over time).
// Phase 2: sequential forward scan; lane j holds alpha[j] (j = lane&15,
// mirrored in the upper half-wave so shuffles stay uniform).
// -------------------------------------------------------------------------
__global__ __launch_bounds__(32) void crf_scan_kernel(
    const float* __restrict__ probs,
    const int*   __restrict__ tags,
    const float* __restrict__ trans,
    const int*   __restrict__ lens,
    float*       __restrict__ ll)
{
    const int b    = blockIdx.x;
    const int lane = threadIdx.x;
    const int j    = lane & 15;
    const int len  = lens[b];
    const int*   tg = tags  + b * SEQ;
    const float* pb = probs + (long)b * SEQ * TAGS;

    // ---- Phase 1: unary + binary partial sums, wave reduction ----
    float u = 0.f, bin = 0.f;
    for (int t = lane; t < SEQ; t += 32) {
        if (t < len) {
            u += pb[t * TAGS + tg[t]];
            if (t >= 1) bin += trans[tg[t - 1] * TAGS + tg[t]];
        }
    }
#pragma unroll
    for (int m = 16; m >= 1; m >>= 1) {
        u   += __shfl_xor(u,   m, 32);
        bin += __shfl_xor(bin, m, 32);
    }

    // ---- Phase 2: forward scan (logsumexp recurrence over T=16 tags) ----
    float tcol[TAGS];
#pragma unroll
    for (int i = 0; i < TAGS; ++i) tcol[i] = trans[i * TAGS + j];

    float alpha = pb[j];  // alpha0 = probs[b, 0, j]
    for (int t = 1; t < SEQ; ++t) {
        if (t >= len) break;  // mask t<len is monotone -> alpha frozen after
        const float emit = pb[t * TAGS + j];
        float vmax = -3.402823466e38f;
        float vals[TAGS];
#pragma unroll
        for (int i = 0; i < TAGS; ++i) {
            float v = __shfl(alpha, i, 32) + tcol[i];
            vals[i] = v;
            vmax = fmaxf(vmax, v);
        }
        float s = 0.f;
#pragma unroll
        for (int i = 0; i < TAGS; ++i) s += expf(vals[i] - vmax);
        alpha = vmax + logf(s) + emit;
    }

    // log_norm = logsumexp over the 16 alphas
    float m2 = -3.402823466e38f;
#pragma unroll
    for (int i = 0; i < TAGS; ++i) m2 = fmaxf(m2, __shfl(alpha, i, 32));
    float s2 = 0.f;
#pragma unroll
    for (int i = 0; i < TAGS; ++i) s2 += expf(__shfl(alpha, i, 32) - m2);
    const float log_norm = m2 + logf(s2);

    if (lane == 0) ll[b] = u + bin - log_norm;
}

// -------------------------------------------------------------------------
// Launch: inputs are text, tags, embed_table, W, b, trans (setup order).
// d_out layout: probs[32768*16] | lens[64] | log_likelihood[64].
// -------------------------------------------------------------------------
extern "C" void kernel_launch(void* const* d_in, const int* in_sizes, int n_in,
                              void* d_out, int out_size, void* d_ws, size_t ws_size,
                              hipStream_t stream) {
    const int*   text  = (const int*)  d_in[0];
    const int*   tags  = (const int*)  d_in[1];
    const float* table = (const float*)d_in[2];
    const float* W     = (const float*)d_in[3];
    const float* bias  = (const float*)d_in[4];
    const float* trans = (const float*)d_in[5];

    float* probs  = (float*)d_out;
    float* lens_f = probs + (long)ROWS * TAGS;  // +524288
    float* ll     = lens_f + BATCH;             // +524352
    int*   lens_i = (int*)d_ws;                 // 64 ints of scratch

    // 2048 tiles of 16 rows, 8 waves (tiles) per 256-thread block
    crf_gemm_kernel<<<ROWS / (16 * 8), 256, 0, stream>>>(text, table, W, bias, probs);
    crf_lens_kernel<<<BATCH, 256, 0, stream>>>(text, lens_f, lens_i);
    crf_scan_kernel<<<BATCH, 32, 0, stream>>>(probs, tags, trans, lens_i, ll);
}